// perfusion_1047972020561
// MI455X (gfx1250) — compile-verified
//
#include <hip/hip_runtime.h>
#include <math.h>

// ---------------------------------------------------------------------------
// out[n] = C0 * param[n,0] * ( S2[n] - 0.5*(log2 cp[n,0] + log2 cp[n,T-1])
//                              - (T-1)*log2 T10[n] )
// S2[n] = sum_t log2 cp[n,t]
// C0 = 0.05 * (RO/KAV) * (1-HLV)/(1-HSV) * (-ln2/TE)
// ---------------------------------------------------------------------------

typedef float v2f __attribute__((ext_vector_type(2)));
typedef float v8f __attribute__((ext_vector_type(8)));

constexpr double TE_d  = 32.0;
constexpr double RO_d  = 1.04;
constexpr double HLV_d = 0.45;
constexpr double HSV_d = 0.25;
constexpr double KAV_d = 1.4;
constexpr double LN2_d = 0.6931471805599453094172321;

constexpr float kC0 =
    (float)(0.05 * (RO_d / KAV_d) * ((1.0 - HLV_d) / (1.0 - HSV_d)) *
            (-(LN2_d) / TE_d));

#define TILE_STRIDE 132   // 16 rows x 128 cols, +4 pad floats -> conflict-free

__global__ __launch_bounds__(256)
void perfusion_rowsum_wmma(const float* __restrict__ param,
                           const float* __restrict__ T10,
                           const float* __restrict__ cp,
                           float* __restrict__ out,
                           int T)
{
    __shared__ float tile[16 * TILE_STRIDE];   // 8448 B staging tile
    __shared__ float wsum[8][16];              // per-wave row sums

    const int tid  = threadIdx.x;
    const int r    = tid >> 4;     // 0..15  : row within tile (load phase)
    const int q    = tid & 15;     // 0..15  : float4 slot within row
    const int w    = tid >> 5;     // 0..7   : wave id
    const int lane = tid & 31;
    const int m    = lane & 15;    // WMMA A-matrix row M
    const int half = lane >> 4;    // 0: K=0,1   1: K=2,3

    const int    row0 = blockIdx.x * 16;
    const float* rowp = cp + (size_t)(row0 + r) * (size_t)T;

    const int nchunks = T >> 7;    // 128 columns per chunk

    v8f acc = {0.f, 0.f, 0.f, 0.f, 0.f, 0.f, 0.f, 0.f};
    v2f ones; ones.x = 1.0f; ones.y = 1.0f;   // B = all-ones 4x16

    // software pipeline: chunk 0 preloaded into registers
    float4 ra = *(const float4*)(rowp + q * 4);
    float4 rb = *(const float4*)(rowp + 64 + q * 4);

    for (int c = 0; c < nchunks; ++c) {
        __syncthreads();                       // prev chunk's readers done
        {
            float* t0 = tile + r * TILE_STRIDE + q * 4;
            float* t1 = t0 + 64;
            ((float2*)t0)[0] = make_float2(ra.x, ra.y);
            ((float2*)t0)[1] = make_float2(ra.z, ra.w);
            ((float2*)t1)[0] = make_float2(rb.x, rb.y);
            ((float2*)t1)[1] = make_float2(rb.z, rb.w);
        }
        __syncthreads();                       // tile visible to all waves

        if (c + 1 < nchunks) {                 // issue next chunk's loads now;
            const float* nxt = rowp + (size_t)(c + 1) * 128;
            ra = *(const float4*)(nxt + q * 4);
            rb = *(const float4*)(nxt + 64 + q * 4);
            if (c + 2 < nchunks)               // and prefetch one more ahead
                __builtin_prefetch(rowp + (size_t)(c + 2) * 128 + q * 4, 0, 0);
        }

        // each wave reduces its private 16-column slice of the tile:
        // A[m][k] = log2(tile[m][w*16 + i*4 + k]),  B = ones  =>
        // D[m][*] += sum_k A[m][k]
        const float* lrow = tile + m * TILE_STRIDE + w * 16 + half * 2;
        #pragma unroll
        for (int i = 0; i < 4; ++i) {
            v2f x = *(const v2f*)(lrow + i * 4);   // ds_load_b64, bank-clean
            v2f a;
            a.x = __log2f(x.x);
            a.y = __log2f(x.y);
            acc = __builtin_amdgcn_wmma_f32_16x16x4_f32(
                      /*neg_a=*/false, a, /*neg_b=*/false, ones,
                      /*c_mod=*/(short)0, acc,
                      /*reuse_a=*/false, /*reuse_b=*/false);
        }
    }

    // D layout: vgpr rr, lanes 0-15 -> M=rr, lanes 16-31 -> M=rr+8;
    // every column N holds the same row sum, so lane m==0 of each half
    // exports 8 rows.
    if (m == 0) {
        #pragma unroll
        for (int rr = 0; rr < 8; ++rr)
            wsum[w][rr + 8 * half] = acc[rr];
    }
    __syncthreads();

    if (tid < 16) {
        float s2 = 0.f;
        #pragma unroll
        for (int ww = 0; ww < 8; ++ww)
            s2 += wsum[ww][tid];

        const int    row = row0 + tid;
        const float* rp  = cp + (size_t)row * (size_t)T;
        const float  l0  = __log2f(rp[0]);
        const float  lN  = __log2f(rp[T - 1]);
        const float  lt  = __log2f(T10[row]);
        const float  val = s2 - 0.5f * (l0 + lN) - (float)(T - 1) * lt;
        out[row] = kC0 * param[row * 3] * val;
    }
}

extern "C" void kernel_launch(void* const* d_in, const int* in_sizes, int n_in,
                              void* d_out, int out_size, void* d_ws, size_t ws_size,
                              hipStream_t stream)
{
    const float* param = (const float*)d_in[0];   // [N,3]
    const float* T10   = (const float*)d_in[1];   // [N,1]
    const float* cp    = (const float*)d_in[2];   // [N,T]
    // d_in[3] (raw_s) is dead in the reference -> never touched (halves traffic)

    const int N = in_sizes[1];           // T10 has N elements
    const int T = in_sizes[2] / N;       // cp is N*T
    float* out = (float*)d_out;

    dim3 grid(N / 16), block(256);
    hipLaunchKernelGGL(perfusion_rowsum_wmma, grid, block, 0, stream,
                       param, T10, cp, out, T);
}